// RDGenerator_64398739636659
// MI455X (gfx1250) — compile-verified
//
#include <hip/hip_runtime.h>
#include <cstdint>

#define HH 1024
#define WW 1024
#define BB 16
#define TT 64

typedef float v2f __attribute__((ext_vector_type(2)));
typedef float v8f __attribute__((ext_vector_type(8)));

__device__ __forceinline__ float sigmoidf_(float x) {
  return 1.0f / (1.0f + __expf(-x));
}
__device__ __forceinline__ float softplusf_(float x) {
  // numerically stable softplus
  return fmaxf(x, 0.0f) + log1pf(__expf(-fabsf(x)));
}
// 1D second-difference (Dirichlet/zero-pad) matrix: tridiag(1, -2, 1)
__device__ __forceinline__ float Lmat(int i, int j) {
  int d = i - j;
  return (d == 0) ? -2.0f : (((d == 1) || (d == -1)) ? 1.0f : 0.0f);
}

// One Gray-Scott step. One wave == one 16x16 tile.
// lap(T) = L*T + T*L  (interior part via WMMA f32 16x16x4), halo fixed up in VALU.
__global__ __launch_bounds__(256) void rd_step_kernel(
    const float* __restrict__ Uin, const float* __restrict__ Vin,
    float* __restrict__ Uout, float* __restrict__ Vout,
    const float* __restrict__ pDu, const float* __restrict__ pDv,
    const float* __restrict__ pF, const float* __restrict__ pk)
{
  constexpr int P = 18;                // 16 + 2 halo, also LDS pitch
  constexpr int TILE_ELEMS = P * P;    // 324
  __shared__ float lds[8][2][TILE_ELEMS];

  const int wave = threadIdx.x >> 5;
  const int lane = threadIdx.x & 31;
  const int b    = blockIdx.z;
  const int r0   = blockIdx.y << 4;                    // tile top row
  const int c0   = ((blockIdx.x << 3) + wave) << 4;    // tile left col

  float* ldsU = &lds[wave][0][0];
  float* ldsV = &lds[wave][1][0];

  const size_t imgOff = (size_t)b * (size_t)(HH * WW);
  const float* gU = Uin + imgOff;
  const float* gV = Vin + imgOff;

  // ---- async copy of 18x18 halo tile (zero outside the image) into LDS ----
#pragma unroll
  for (int it = 0; it < 11; ++it) {
    int j = lane + (it << 5);
    if (j < TILE_ELEMS) {
      int r  = j / P;
      int c  = j - r * P;
      int gr = r0 - 1 + r;
      int gc = c0 - 1 + c;
      if (((unsigned)gr < (unsigned)HH) && ((unsigned)gc < (unsigned)WW)) {
        unsigned off = (((unsigned)gr << 10) + (unsigned)gc) << 2;  // byte offset
        unsigned lu  = (unsigned)(size_t)&ldsU[j];
        unsigned lv  = (unsigned)(size_t)&ldsV[j];
        asm volatile("global_load_async_to_lds_b32 %0, %1, %2"
                     :: "v"(lu), "v"(off), "s"(gU) : "memory");
        asm volatile("global_load_async_to_lds_b32 %0, %1, %2"
                     :: "v"(lv), "v"(off), "s"(gV) : "memory");
      } else {
        ldsU[j] = 0.0f;
        ldsV[j] = 0.0f;
      }
    }
  }
  asm volatile("s_wait_asynccnt 0" ::: "memory");

  // ---- Laplacian via WMMA: acc = L*T + T*L (interior contribution) ----
  const int h  = lane >> 4;   // lane half
  const int nn = lane & 15;   // N (or M for A operands)

  v8f accU = {0.f, 0.f, 0.f, 0.f, 0.f, 0.f, 0.f, 0.f};
  v8f accV = {0.f, 0.f, 0.f, 0.f, 0.f, 0.f, 0.f, 0.f};

  // product 1: A = L (constant), B = tile   => row stencil
#pragma unroll
  for (int k = 0; k < 4; ++k) {
    const int k0 = (k << 2) + (h << 1);  // K slice: VGPR0 -> k0, VGPR1 -> k0+1
    v2f aL; aL.x = Lmat(nn, k0);         aL.y = Lmat(nn, k0 + 1);
    v2f bU; bU.x = ldsU[(k0 + 1) * P + nn + 1]; bU.y = ldsU[(k0 + 2) * P + nn + 1];
    v2f bV; bV.x = ldsV[(k0 + 1) * P + nn + 1]; bV.y = ldsV[(k0 + 2) * P + nn + 1];
    accU = __builtin_amdgcn_wmma_f32_16x16x4_f32(false, aL, false, bU, (short)0, accU, false, false);
    accV = __builtin_amdgcn_wmma_f32_16x16x4_f32(false, aL, false, bV, (short)0, accV, false, false);
  }
  // product 2: A = tile, B = L (constant)   => column stencil
#pragma unroll
  for (int k = 0; k < 4; ++k) {
    const int k0 = (k << 2) + (h << 1);
    v2f bL; bL.x = Lmat(k0, nn);         bL.y = Lmat(k0 + 1, nn);
    v2f aU; aU.x = ldsU[(nn + 1) * P + k0 + 1]; aU.y = ldsU[(nn + 1) * P + k0 + 2];
    v2f aV; aV.x = ldsV[(nn + 1) * P + k0 + 1]; aV.y = ldsV[(nn + 1) * P + k0 + 2];
    accU = __builtin_amdgcn_wmma_f32_16x16x4_f32(false, aU, false, bL, (short)0, accU, false, false);
    accV = __builtin_amdgcn_wmma_f32_16x16x4_f32(false, aV, false, bL, (short)0, accV, false, false);
  }

  // ---- reaction + sigmoid + store (C/D layout: VGPR r -> M = r + 8*h, N = nn) ----
  const float Du = softplusf_(pDu[0]);
  const float Dv = softplusf_(pDv[0]);
  const float Ff = 0.1f * sigmoidf_(pF[0]);
  const float kp = 0.1f * sigmoidf_(pk[0]);

  float* oU = Uout + imgOff;
  float* oV = Vout + imgOff;

#pragma unroll
  for (int r = 0; r < 8; ++r) {
    const int m = r + (h << 3);
    float lu = accU[r];
    float lv = accV[r];
    // halo corrections at tile edges
    if (r == 0) {  // M == 0 possible only for h == 0
      lu += (h == 0) ? ldsU[nn + 1] : 0.0f;
      lv += (h == 0) ? ldsV[nn + 1] : 0.0f;
    }
    if (r == 7) {  // M == 15 possible only for h == 1
      lu += (h == 1) ? ldsU[17 * P + nn + 1] : 0.0f;
      lv += (h == 1) ? ldsV[17 * P + nn + 1] : 0.0f;
    }
    lu += (nn == 0)  ? ldsU[(m + 1) * P]      : 0.0f;
    lv += (nn == 0)  ? ldsV[(m + 1) * P]      : 0.0f;
    lu += (nn == 15) ? ldsU[(m + 1) * P + 17] : 0.0f;
    lv += (nn == 15) ? ldsV[(m + 1) * P + 17] : 0.0f;

    const float u   = ldsU[(m + 1) * P + nn + 1];
    const float v   = ldsV[(m + 1) * P + nn + 1];
    const float uvv = u * v * v;
    const float du  = Du * lu - uvv + Ff * (1.0f - u);
    const float dv  = Dv * lv + uvv - (Ff + kp) * v;

    const size_t o = (size_t)(r0 + m) * WW + (size_t)(c0 + nn);
    oU[o] = sigmoidf_(u + du);   // DT == 1.0
    oV[o] = sigmoidf_(v + dv);
  }
}

extern "C" void kernel_launch(void* const* d_in, const int* in_sizes, int n_in,
                              void* d_out, int out_size, void* d_ws, size_t ws_size,
                              hipStream_t stream) {
  (void)in_sizes; (void)n_in; (void)out_size; (void)ws_size;

  const float* U0  = (const float*)d_in[0];
  const float* V0  = (const float*)d_in[1];
  const float* pDu = (const float*)d_in[2];
  const float* pDv = (const float*)d_in[3];
  const float* pF  = (const float*)d_in[4];
  const float* pk  = (const float*)d_in[5];

  const size_t N = (size_t)BB * HH * WW;
  float* uA = (float*)d_ws;        // U ping
  float* uB = uA + N;              // U pong
  float* vA = uB + N;              // V ping
  float* vB = (float*)d_out;       // V pong == output (step 63 is odd -> writes here)

  dim3 grid(WW / 128, HH / 16, BB);   // 8 x 64 x 16 blocks; 8 waves (tiles) per block
  dim3 block(256);

  for (int t = 0; t < TT; ++t) {
    const float* Ui = (t == 0) ? U0 : ((t & 1) ? uA : uB);
    const float* Vi = (t == 0) ? V0 : ((t & 1) ? vA : vB);
    float* Uo = (t & 1) ? uB : uA;
    float* Vo = (t & 1) ? vB : vA;
    hipLaunchKernelGGL(rd_step_kernel, grid, block, 0, stream,
                       Ui, Vi, Uo, Vo, pDu, pDv, pF, pk);
  }
}